// Tokenizer_5282809774843
// MI455X (gfx1250) — compile-verified
//
#include <hip/hip_runtime.h>
#include <hip/hip_bf16.h>
#include <stdint.h>

// Problem constants (from reference): B=32, L=1024, A=1024, E=256, V=4096
#define NTOK   32768      // B*L
#define ADIM   1024
#define EDIM   256
#define VDIM   4096

#define KC     16         // GEMM K-chunk staged in LDS
#define SA     28         // A-tile LDS row stride (floats): 112B (16B mult), 28*l mod 64 distinct over 16 lanes
#define SBB    136        // B-tile LDS row stride (floats): 544B (16B mult), 2*136 mod 64 = 16 -> lane halves disjoint
#define LDS_SP 268        // dist emb-tile row stride: 1072B (16B mult), 12*n mod 64 distinct; halves disjoint

typedef __attribute__((ext_vector_type(2))) float v2f;
typedef __attribute__((ext_vector_type(8))) float v8f;
typedef int i32x4 __attribute__((vector_size(4 * sizeof(int))));   // matches builtin param type

// ---------------------------------------------------------------------------
// Async global->LDS copy (CDNA5 GLOBAL_LOAD_ASYNC_TO_LDS_B128, ASYNCcnt),
// with graceful fallback to a synchronous copy if the builtin is absent.
// Builtin signature (from hipcc diagnostic): param0 = i32x4 AS(1)*, then
// i32x4 AS(3)*, const int offset, const int cpol.
// ---------------------------------------------------------------------------
#if __has_builtin(__builtin_amdgcn_global_load_async_to_lds_b128)
#define HAVE_ASYNC_LDS 1
#else
#define HAVE_ASYNC_LDS 0
#endif

__device__ __forceinline__ void cp_b128(float* lds_dst, const float* gsrc) {
#if HAVE_ASYNC_LDS
    __builtin_amdgcn_global_load_async_to_lds_b128(
        (__attribute__((address_space(1))) i32x4*)(uintptr_t)gsrc,
        (__attribute__((address_space(3))) i32x4*)(uintptr_t)lds_dst,
        /*offset=*/0, /*cpol=*/0);
#else
    *(float4*)lds_dst = *(const float4*)gsrc;
#endif
}

__device__ __forceinline__ void cp_commit_wait() {
#if HAVE_ASYNC_LDS
#if __has_builtin(__builtin_amdgcn_s_wait_asynccnt)
    __builtin_amdgcn_s_wait_asynccnt(0);
#else
    asm volatile("s_wait_asynccnt 0" ::: "memory");
#endif
#endif
}

// ---------------------------------------------------------------------------
// Kernel 1: embsq[v] = sum_e emb[v][e]^2   (wave per row, shuffle reduce)
// ---------------------------------------------------------------------------
__global__ void __launch_bounds__(256) embsq_kernel(const float* __restrict__ emb,
                                                    float* __restrict__ embsq) {
    int v    = blockIdx.x * 8 + (threadIdx.x >> 5);
    int lane = threadIdx.x & 31;
    const float* row = emb + (size_t)v * EDIM;
    float s = 0.0f;
#pragma unroll
    for (int i = 0; i < EDIM / 32; ++i) {
        float t = row[lane + 32 * i];
        s += t * t;
    }
#pragma unroll
    for (int m = 16; m >= 1; m >>= 1) s += __shfl_xor(s, m, 32);
    if (lane == 0) embsq[v] = s;
}

// ---------------------------------------------------------------------------
// Kernel 2/5: C[M x NCOLS] = A[M x K] * B[K x NCOLS], row-major fp32.
// Block = 256 thr (8 waves) -> C tile 128x128. Wave = 2 M-tiles x 4 N-tiles
// (B fragments reused across the two M-tiles). K chunked by KC through
// double-buffered LDS fed by async b128 copies; chunk k+1 streams in while
// chunk k is consumed by V_WMMA_F32_16X16X4_F32.
//
// WMMA f32 16x16x4 lane layout (ISA 7.12.2):
//   A: lane(m<16) v0=A[m][k0],v1=A[m][k0+1]; lanes 16-31 -> k0+2,k0+3
//   B: lane(n<16) v0=B[k0][n],v1=B[k0+1][n]; lanes 16-31 -> k0+2,k0+3
//   C: reg r, lanes 0-15 -> (M=r,N=lane); lanes 16-31 -> (M=r+8,N=lane-16)
// ---------------------------------------------------------------------------
template <int K, int NCOLS>
__global__ void __launch_bounds__(256) wmma_gemm(const float* __restrict__ Am,
                                                 const float* __restrict__ Bm,
                                                 float* __restrict__ Cm) {
    __shared__ float Abuf[2][128 * SA];   // 128 rows x KC cols (padded)
    __shared__ float Bbuf[2][KC * SBB];   // KC rows x 128 cols (padded)

    const int tid   = threadIdx.x;
    const int lane  = tid & 31;
    const int l15   = lane & 15;
    const int khalf = (lane >> 4) << 1;    // 0 | 2
    const int wave  = tid >> 5;
    const int wm    = wave & 3;            // rows wm*32 .. +31 (two 16-row tiles)
    const int wn    = wave >> 2;           // cols wn*64 .. +63 (four 16-col tiles)
    const int mblk  = blockIdx.x * 128;
    const int nblk  = blockIdx.y * 128;

    v8f acc[2][4];
#pragma unroll
    for (int u = 0; u < 2; ++u)
#pragma unroll
        for (int t = 0; t < 4; ++t) acc[u][t] = (v8f){0,0,0,0,0,0,0,0};

    // Cooperative chunk fetch: A 128x16 = 512 float4, B 16x128 = 512 float4
    auto issue = [&](int buf, int kc) {
#pragma unroll
        for (int j = 0; j < 2; ++j) {                 // A: 2 float4 per thread
            int i = tid + j * 256;
            int r = i >> 2, c = (i & 3) << 2;
            cp_b128(&Abuf[buf][r * SA + c],
                    Am + (size_t)(mblk + r) * K + (kc + c));
        }
#pragma unroll
        for (int j = 0; j < 2; ++j) {                 // B: 2 float4 per thread
            int i = tid + j * 256;
            int r = i >> 5, c = (i & 31) << 2;
            cp_b128(&Bbuf[buf][r * SBB + c],
                    Bm + (size_t)(kc + r) * NCOLS + (nblk + c));
        }
    };

    issue(0, 0);
    cp_commit_wait();
    __syncthreads();

    constexpr int NCH = K / KC;
    for (int ch = 0; ch < NCH; ++ch) {
        const int cur = ch & 1;
        if (ch + 1 < NCH) issue(cur ^ 1, (ch + 1) * KC);   // overlap next chunk

        const float* a0p   = &Abuf[cur][(wm * 32 + l15) * SA + khalf];
        const float* a1p   = a0p + 16 * SA;
        const float* bbase = &Bbuf[cur][khalf * SBB + wn * 64 + l15];
#pragma unroll
        for (int ks = 0; ks < KC / 4; ++ks) {
            v2f a0 = *(const v2f*)(a0p + ks * 4);
            v2f a1 = *(const v2f*)(a1p + ks * 4);
            const float* bp = bbase + ks * 4 * SBB;
#pragma unroll
            for (int t = 0; t < 4; ++t) {
                v2f b;
                b.x = bp[t * 16];
                b.y = bp[t * 16 + SBB];
                acc[0][t] = __builtin_amdgcn_wmma_f32_16x16x4_f32(
                    false, a0, false, b, (short)0, acc[0][t], false, false);
                acc[1][t] = __builtin_amdgcn_wmma_f32_16x16x4_f32(
                    false, a1, false, b, (short)0, acc[1][t], false, false);
            }
        }
        cp_commit_wait();
        __syncthreads();
    }

    const int rbase = mblk + wm * 32 + ((lane >> 4) << 3);
#pragma unroll
    for (int u = 0; u < 2; ++u) {
#pragma unroll
        for (int t = 0; t < 4; ++t) {
            const int col = nblk + wn * 64 + t * 16 + l15;
#pragma unroll
            for (int r = 0; r < 8; ++r) {
                Cm[(size_t)(rbase + u * 16 + r) * NCOLS + col] = acc[u][t][r];
            }
        }
    }
}

// ---------------------------------------------------------------------------
// Kernel 3: tokens[n] = argmin_v ( embsq[v] - 2 * <z_n, emb_v> )
// Wave owns a 16-token tile; all 64 A-fragments of z preloaded into VGPRs.
// Embedding tiles (16x256) double-buffered in LDS via async b128 copies.
// ---------------------------------------------------------------------------
__global__ void __launch_bounds__(256) dist_argmin(const float* __restrict__ z,
                                                   const float* __restrict__ emb,
                                                   const float* __restrict__ embsq,
                                                   int* __restrict__ tokens) {
    __shared__ float btile[2][16 * LDS_SP];

    const int tid   = threadIdx.x;
    const int wave  = tid >> 5;
    const int lane  = tid & 31;
    const int l15   = lane & 15;
    const int khalf = (lane >> 4) << 1;
    const int m0    = (blockIdx.x * 8 + wave) * 16;

    // Preload all A-matrix fragments of this 16 x 256 z-tile (64 x float2)
    v2f afr[EDIM / 4];
    const float* zr = z + (size_t)(m0 + l15) * EDIM;
#pragma unroll
    for (int ks = 0; ks < EDIM / 4; ++ks)
        afr[ks] = *(const v2f*)(zr + 4 * ks + khalf);

    float minv[8];
    int   mini[8];
#pragma unroll
    for (int r = 0; r < 8; ++r) { minv[r] = 3.4e38f; mini[r] = 0; }

    auto issue = [&](int buf, int v0) {
#pragma unroll
        for (int j = 0; j < 4; ++j) {     // 16x256 = 1024 float4 / 256 thr
            int i  = tid + j * 256;
            int rv = i >> 6;
            int cc = (i & 63) << 2;
            cp_b128(&btile[buf][rv * LDS_SP + cc],
                    emb + (size_t)(v0 + rv) * EDIM + cc);
        }
    };

    issue(0, 0);
    cp_commit_wait();
    __syncthreads();

    for (int vt = 0; vt < VDIM / 16; ++vt) {
        const int cur = vt & 1;
        if (vt + 1 < VDIM / 16) issue(cur ^ 1, (vt + 1) * 16);

        v8f acc = (v8f){0,0,0,0,0,0,0,0};
#pragma unroll
        for (int ks = 0; ks < EDIM / 4; ++ks) {
            // B[k][n] = emb[v0+n][k] -> contiguous float2 in LDS tile row
            v2f b = *(const v2f*)(&btile[cur][l15 * LDS_SP + 4 * ks + khalf]);
            acc = __builtin_amdgcn_wmma_f32_16x16x4_f32(
                false, afr[ks], false, b, (short)0, acc, false, false);
        }

        const int   v0   = vt * 16;
        const float sq   = embsq[v0 + l15];
        const int   vidx = v0 + l15;
#pragma unroll
        for (int r = 0; r < 8; ++r) {
            float d = sq - 2.0f * acc[r];    // ||z||^2 constant per row: omitted
            if (d < minv[r]) { minv[r] = d; mini[r] = vidx; }  // strict < keeps lowest v
        }
        cp_commit_wait();
        __syncthreads();
    }

    // Reduce across the 16 lanes holding each row (xor masks stay in-half)
#pragma unroll
    for (int r = 0; r < 8; ++r) {
        float mv = minv[r];
        int   mi = mini[r];
#pragma unroll
        for (int off = 8; off >= 1; off >>= 1) {
            float ov = __shfl_xor(mv, off, 32);
            int   oi = __shfl_xor(mi, off, 32);
            if (ov < mv || (ov == mv && oi < mi)) { mv = ov; mi = oi; }
        }
        if (l15 == 0) tokens[m0 + r + ((lane >> 4) << 3)] = mi;
    }
}

// ---------------------------------------------------------------------------
// Kernel 4: z_q[n][e] = emb[tokens[n]][e]  (block per token, coalesced copy)
// ---------------------------------------------------------------------------
__global__ void __launch_bounds__(256) gather_zq(const float* __restrict__ emb,
                                                 const int* __restrict__ tokens,
                                                 float* __restrict__ zq) {
    int n = blockIdx.x;
    int e = threadIdx.x;           // EDIM == 256 == blockDim.x
    zq[(size_t)n * EDIM + e] = emb[(size_t)tokens[n] * EDIM + e];
}

// ---------------------------------------------------------------------------
// Launcher
// inputs : x[32768x1024], W_enc[1024x256], W_dec[256x1024], embedding[4096x256]
// output : z | z_q | reconstructions (fp32, concat flat)
// ws     : embsq[4096] f32, tokens[32768] i32
// ---------------------------------------------------------------------------
extern "C" void kernel_launch(void* const* d_in, const int* in_sizes, int n_in,
                              void* d_out, int out_size, void* d_ws, size_t ws_size,
                              hipStream_t stream) {
    const float* x     = (const float*)d_in[0];
    const float* W_enc = (const float*)d_in[1];
    const float* W_dec = (const float*)d_in[2];
    const float* emb   = (const float*)d_in[3];

    float* z     = (float*)d_out;
    float* zq    = z + (size_t)NTOK * EDIM;
    float* recon = zq + (size_t)NTOK * EDIM;

    float* embsq  = (float*)d_ws;
    int*   tokens = (int*)((char*)d_ws + VDIM * sizeof(float));

    // 1) ||emb_v||^2
    embsq_kernel<<<VDIM / 8, 256, 0, stream>>>(emb, embsq);

    // 2) z = x @ W_enc          (M=32768, K=1024, N=256)
    wmma_gemm<ADIM, EDIM><<<dim3(NTOK / 128, EDIM / 128), 256, 0, stream>>>(x, W_enc, z);

    // 3) tokens = argmin_v dist (M=32768, K=256, V=4096)
    dist_argmin<<<NTOK / 128, 256, 0, stream>>>(z, emb, embsq, tokens);

    // 4) z_q = emb[tokens]
    gather_zq<<<NTOK, 256, 0, stream>>>(emb, tokens, zq);

    // 5) recon = z_q @ W_dec    (M=32768, K=256, N=1024)
    wmma_gemm<EDIM, ADIM><<<dim3(NTOK / 128, ADIM / 128), 256, 0, stream>>>(zq, W_dec, recon);
}